// GATConv_56916906607109
// MI455X (gfx1250) — compile-verified
//
#include <hip/hip_runtime.h>
#include <math.h>

#define IN_F 256
#define OUT_F 128
#define NEG_SLOPE 0.2f

typedef __attribute__((ext_vector_type(2))) float v2f;
typedef __attribute__((ext_vector_type(8))) float v8f;

// ---------------------------------------------------------------------------
// Init: zero rst + denom, set m = -inf
// ---------------------------------------------------------------------------
__global__ __launch_bounds__(256) void gat_init(float* __restrict__ rst,
                                                float* __restrict__ m,
                                                float* __restrict__ denom,
                                                int n_rst, int n_nodes) {
    int i = blockIdx.x * blockDim.x + threadIdx.x;
    int stride = gridDim.x * blockDim.x;
    for (int j = i; j < n_rst; j += stride) rst[j] = 0.0f;
    for (int j = i; j < n_nodes; j += stride) {
        m[j] = -__builtin_inff();
        denom[j] = 0.0f;
    }
}

// ---------------------------------------------------------------------------
// GEMM: h[n,d] = sum_k feat[n,k] * W[d,k]   via V_WMMA_F32_16X16X4_F32
// Block = 256 threads = 8 waves. Block owns 16 node rows (A tile in LDS);
// wave w owns output columns [16w, 16w+16). K loop: 256 / 4 = 64 WMMAs/wave.
// Requires n_nodes % 16 == 0 (holds: 50000 = 3125*16); EXEC all-ones.
// ---------------------------------------------------------------------------
__global__ __launch_bounds__(256) void gat_gemm(const float* __restrict__ feat,
                                                const float* __restrict__ W,
                                                float* __restrict__ h) {
    __shared__ float Atile[16 * IN_F];  // 16 KB

    const int tid = threadIdx.x;
    const int rowBase = blockIdx.x * 16;

    // Stage 16 x 256 f32 of feat into LDS: 1024 float4, 4 per thread.
    {
        const float4* src = (const float4*)(feat + (size_t)rowBase * IN_F);
        float4* dstp = (float4*)Atile;
#pragma unroll
        for (int j = 0; j < 4; ++j) dstp[tid + 256 * j] = src[tid + 256 * j];
    }
    __syncthreads();

    const int lane  = tid & 31;
    const int wave  = tid >> 5;
    const int colBase = wave * 16;
    const int l16   = lane & 15;
    const int khalf = (lane >> 4) * 2;  // lanes 0-15 -> K+0,K+1 ; lanes 16-31 -> K+2,K+3

    const float* Arow = Atile + l16 * IN_F;                    // A: M = lane%16
    const float* Brow = W + (size_t)(colBase + l16) * IN_F;    // B: N = lane%16 (B[k][n]=W[n][k])

    v8f c = {};
#pragma unroll 4
    for (int k = 0; k < IN_F; k += 4) {
        v2f a, b;
        a.x = Arow[k + khalf];
        a.y = Arow[k + khalf + 1];
        b.x = Brow[k + khalf];
        b.y = Brow[k + khalf + 1];
        // (neg_a, A, neg_b, B, c_mod, C, reuse_a, reuse_b)
        c = __builtin_amdgcn_wmma_f32_16x16x4_f32(false, a, false, b,
                                                  (short)0, c, false, false);
    }

    // C/D layout: VGPR r -> M = r (lanes 0-15) / r+8 (lanes 16-31), N = lane%16
    const int mOff = (lane >> 4) * 8;
    float* out = h + (size_t)(rowBase + mOff) * OUT_F + colBase + l16;
#pragma unroll
    for (int r = 0; r < 8; ++r) out[(size_t)r * OUT_F] = c[r];
}

// ---------------------------------------------------------------------------
// Attention logits: el[n] = h[n,:].attn_l ; er[n] = h[n,:].attn_r
// One wave (32 lanes) per node; 4 columns per lane; shfl reduction.
// ---------------------------------------------------------------------------
__global__ __launch_bounds__(256) void gat_attn(const float* __restrict__ h,
                                                const float* __restrict__ attn_l,
                                                const float* __restrict__ attn_r,
                                                float* __restrict__ el,
                                                float* __restrict__ er,
                                                int n_nodes) {
    const int lane = threadIdx.x & 31;
    const int node = blockIdx.x * 8 + (threadIdx.x >> 5);
    if (node >= n_nodes) return;
    const float* hr = h + (size_t)node * OUT_F;
    float sl = 0.0f, sr = 0.0f;
#pragma unroll
    for (int j = 0; j < 4; ++j) {
        int cidx = lane + 32 * j;
        float v = hr[cidx];
        sl += v * attn_l[cidx];
        sr += v * attn_r[cidx];
    }
#pragma unroll
    for (int off = 16; off > 0; off >>= 1) {
        sl += __shfl_xor(sl, off, 32);
        sr += __shfl_xor(sr, off, 32);
    }
    if (lane == 0) { el[node] = sl; er[node] = sr; }
}

// Ordered-int float atomic max (monotone bit trick), init value must be -inf.
__device__ inline void atomicMaxF(float* addr, float val) {
    if (val >= 0.0f)
        atomicMax((int*)addr, __float_as_int(val));
    else
        atomicMin((unsigned int*)addr, __float_as_uint(val));
}

// ---------------------------------------------------------------------------
// Edge pass 1: e = leaky_relu(el[src] + er[dst]); segment-max into m[dst]
// ---------------------------------------------------------------------------
__global__ __launch_bounds__(256) void gat_edge1(const int* __restrict__ src,
                                                 const int* __restrict__ dst,
                                                 const float* __restrict__ el,
                                                 const float* __restrict__ er,
                                                 float* __restrict__ ebuf,
                                                 float* __restrict__ m,
                                                 int n_edges) {
    int i = blockIdx.x * blockDim.x + threadIdx.x;
    if (i >= n_edges) return;
    float e = el[src[i]] + er[dst[i]];
    e = (e > 0.0f) ? e : NEG_SLOPE * e;
    ebuf[i] = e;
    atomicMaxF(&m[dst[i]], e);
}

// ---------------------------------------------------------------------------
// Edge pass 2: ex = exp(e - m[dst]) (overwrites ebuf); segment-sum into denom
// ---------------------------------------------------------------------------
__global__ __launch_bounds__(256) void gat_edge2(const int* __restrict__ dst,
                                                 float* __restrict__ ebuf,
                                                 const float* __restrict__ m,
                                                 float* __restrict__ denom,
                                                 int n_edges) {
    int i = blockIdx.x * blockDim.x + threadIdx.x;
    if (i >= n_edges) return;
    int d = dst[i];
    float ex = expf(ebuf[i] - m[d]);
    ebuf[i] = ex;
    atomicAdd(&denom[d], ex);
}

// ---------------------------------------------------------------------------
// Edge pass 3: rst[dst] += (ex/denom[dst]) * h[src]
// One wave per edge; lane l handles columns l, l+32, l+64, l+96 (coalesced).
// ---------------------------------------------------------------------------
__global__ __launch_bounds__(256) void gat_edge3(const int* __restrict__ src,
                                                 const int* __restrict__ dst,
                                                 const float* __restrict__ ebuf,
                                                 const float* __restrict__ denom,
                                                 const float* __restrict__ h,
                                                 float* __restrict__ rst,
                                                 int n_edges) {
    const int lane = threadIdx.x & 31;
    const int e = blockIdx.x * 8 + (threadIdx.x >> 5);
    if (e >= n_edges) return;
    const int s = src[e], d = dst[e];
    const float alpha = ebuf[e] / fmaxf(denom[d], 1e-9f);
    const float* hs = h + (size_t)s * OUT_F;
    float* rd = rst + (size_t)d * OUT_F;
#pragma unroll
    for (int j = 0; j < 4; ++j) {
        int cidx = lane + 32 * j;
        atomicAdd(&rd[cidx], alpha * hs[cidx]);
    }
}

// ---------------------------------------------------------------------------
extern "C" void kernel_launch(void* const* d_in, const int* in_sizes, int n_in,
                              void* d_out, int out_size, void* d_ws, size_t ws_size,
                              hipStream_t stream) {
    const float* feat   = (const float*)d_in[0];
    const float* W      = (const float*)d_in[1];
    const float* attn_l = (const float*)d_in[2];
    const float* attn_r = (const float*)d_in[3];
    const int*   src    = (const int*)d_in[4];
    const int*   dst    = (const int*)d_in[5];
    float*       rst    = (float*)d_out;

    const int n_nodes = in_sizes[0] / IN_F;   // 50000 (multiple of 16)
    const int n_edges = in_sizes[4];          // 1600000

    // Workspace layout (floats): h | el | er | m | denom | ebuf  (~32.8 MB)
    float* h     = (float*)d_ws;
    float* el    = h + (size_t)n_nodes * OUT_F;
    float* er    = el + n_nodes;
    float* m     = er + n_nodes;
    float* denom = m + n_nodes;
    float* ebuf  = denom + n_nodes;

    const int n_rst = n_nodes * OUT_F;

    gat_init<<<1024, 256, 0, stream>>>(rst, m, denom, n_rst, n_nodes);

    gat_gemm<<<n_nodes / 16, 256, 0, stream>>>(feat, W, h);

    gat_attn<<<(n_nodes + 7) / 8, 256, 0, stream>>>(h, attn_l, attn_r, el, er, n_nodes);

    const int eb = (n_edges + 255) / 256;
    gat_edge1<<<eb, 256, 0, stream>>>(src, dst, el, er, ebuf, m, n_edges);
    gat_edge2<<<eb, 256, 0, stream>>>(dst, ebuf, m, denom, n_edges);
    gat_edge3<<<(n_edges + 7) / 8, 256, 0, stream>>>(src, dst, ebuf, denom, h, rst, n_edges);
}